// STA_80831284511446
// MI455X (gfx1250) — compile-verified
//
#include <hip/hip_runtime.h>

#define NN 8192
#define LL 4

typedef float v2f __attribute__((ext_vector_type(2)));
typedef float v4f __attribute__((ext_vector_type(4)));
typedef float v8f __attribute__((ext_vector_type(8)));

// ---------------------------------------------------------------------------
// ws layout (floats): [0..15] bounds (2 used), then 6 planes of 4*8192
// ---------------------------------------------------------------------------

__global__ void k_init(float* bounds) {
  if (threadIdx.x < 2) bounds[threadIdx.x] = 0.0f;
}

// Xe = W1x @ S, Se = W1s @ X, plus per-branch bound m = max_i |x_i|^2
__global__ __launch_bounds__(128) void k_proj(
    const float* __restrict__ X, const float* __restrict__ S,
    const float* __restrict__ W1x, const float* __restrict__ W1s,
    float* __restrict__ Xe, float* __restrict__ Se,
    float* __restrict__ bounds) {
  const int n = blockIdx.x * 128 + threadIdx.x;
  float x[LL], s[LL];
#pragma unroll
  for (int l = 0; l < LL; ++l) { x[l] = X[l * NN + n]; s[l] = S[l * NN + n]; }
  float nx = 0.f, ns = 0.f;
#pragma unroll
  for (int l = 0; l < LL; ++l) {
    float xe = 0.f, se = 0.f;
#pragma unroll
    for (int k = 0; k < LL; ++k) {
      xe = fmaf(W1x[l * LL + k], s[k], xe);
      se = fmaf(W1s[l * LL + k], x[k], se);
    }
    Xe[l * NN + n] = xe;
    Se[l * NN + n] = se;
    nx = fmaf(xe, xe, nx);
    ns = fmaf(se, se, ns);
  }
#pragma unroll
  for (int off = 16; off > 0; off >>= 1) {
    nx = fmaxf(nx, __shfl_xor(nx, off));
    ns = fmaxf(ns, __shfl_xor(ns, off));
  }
  if ((threadIdx.x & 31) == 0) {
    // normsq >= 0, so uint compare is order-preserving
    atomicMax((unsigned int*)&bounds[0], __float_as_uint(nx));
    atomicMax((unsigned int*)&bounds[1], __float_as_uint(ns));
  }
}

// Fused column-softmax attention: per wave, 16 columns j; stream i in chunks
// of 16. G^T tile = A(16x4: Xe[:,j-tile]^T) x B(4x16: Xe[:,i-chunk]) via
// V_WMMA_F32_16X16X4_F32.  C layout: vgpr r -> row j=r (lanes 0-15) / j=r+8
// (lanes 16-31); column i = lane&15.
// No LDS / no cross-lane ops in the hot loop: each lane loads all 4 rows of
// its i-column (L2-resident 128KB buffer) and builds the B operand by select.
// exp via raw v_exp_f32 (args are <= 0 after bound subtraction; FTZ of the
// softmax tail below 2^-126 is harmless).
__global__ __launch_bounds__(128) void k_attn(
    const float* __restrict__ XeB, const float* __restrict__ SeB,
    float* __restrict__ XaB, float* __restrict__ SaB,
    const float* __restrict__ bounds) {
  const int branch = blockIdx.y;
  const float* E = branch ? SeB : XeB;
  float* A = branch ? SaB : XaB;
  const float LOG2E = 1.44269504088896340736f;
  const float nb = -bounds[branch] * LOG2E;  // -m * log2(e)

  const int wave = threadIdx.x >> 5;
  const int lane = threadIdx.x & 31;
  const int il = lane & 15;
  const int hi = lane >> 4;           // 0: low half (l=0,1), 1: high half (l=2,3)
  const int rsel = hi ? 2 : 0;
  const int j0 = (blockIdx.x * 4 + wave) * 16;

  // A operand (16x4, M=j, K=l): lane<16 -> v0=K0,v1=K1 ; lane>=16 -> v0=K2,v1=K3
  v2f a;
  a.x = E[rsel * NN + j0 + il];
  a.y = E[(rsel + 1) * NN + j0 + il];

  float den[8], num0[8], num1[8], num2[8], num3[8];
#pragma unroll
  for (int r = 0; r < 8; ++r) {
    den[r] = 0.f; num0[r] = 0.f; num1[r] = 0.f; num2[r] = 0.f; num3[r] = 0.f;
  }

  for (int i0 = 0; i0 < NN; i0 += 16) {
    const float e0 = E[0 * NN + i0 + il];
    const float e1 = E[1 * NN + i0 + il];
    const float e2 = E[2 * NN + i0 + il];
    const float e3 = E[3 * NN + i0 + il];
    // B operand (4x16, K=l, N=i): v0 = K0 (low lanes) / K2 (high lanes), etc.
    v2f b;
    b.x = hi ? e2 : e0;
    b.y = hi ? e3 : e1;
    v8f c = {0.f, 0.f, 0.f, 0.f, 0.f, 0.f, 0.f, 0.f};
    v8f d = __builtin_amdgcn_wmma_f32_16x16x4_f32(
        false, a, false, b, (short)0, c, false, false);

#pragma unroll
    for (int r = 0; r < 8; ++r) {
      // exp(G - m) == exp2(G*log2e - m*log2e): one v_pk_fma half + v_exp_f32
      float p = __builtin_amdgcn_exp2f(fmaf(d[r], LOG2E, nb));
      den[r] += p;
      num0[r] = fmaf(p, e0, num0[r]);
      num1[r] = fmaf(p, e1, num1[r]);
      num2[r] = fmaf(p, e2, num2[r]);
      num3[r] = fmaf(p, e3, num3[r]);
    }
  }

  // Sum over the 16 lanes of each half (they hold different i, same j rows)
#pragma unroll
  for (int off = 1; off < 16; off <<= 1) {
#pragma unroll
    for (int r = 0; r < 8; ++r) {
      den[r]  += __shfl_xor(den[r],  off);
      num0[r] += __shfl_xor(num0[r], off);
      num1[r] += __shfl_xor(num1[r], off);
      num2[r] += __shfl_xor(num2[r], off);
      num3[r] += __shfl_xor(num3[r], off);
    }
  }

  if (il == 0) {
    const int jb = j0 + (hi ? 8 : 0);
#pragma unroll
    for (int r = 0; r < 8; ++r) {
      const float inv = 1.0f / den[r];
      const int j = jb + r;
      A[0 * NN + j] = num0[r] * inv;
      A[1 * NN + j] = num1[r] * inv;
      A[2 * NN + j] = num2[r] * inv;
      A[3 * NN + j] = num3[r] * inv;
    }
  }
}

// Memory-bound GEMM: (8192x8192) @ (8192x4) for both branches fused so W_2x
// streams from HBM exactly once.  One wave per output row, b128 coalesced,
// non-temporal on the 256MB weight stream so the L2-resident RHS survives.
// RHS[k,c] = XaFlat[k*4+c] (the reshape is a pure flat reinterpret).
__global__ __launch_bounds__(128) void k_gemm2(
    const float* __restrict__ W2, const float* __restrict__ XaF,
    const float* __restrict__ SaF, float* __restrict__ MxF,
    float* __restrict__ MsF) {
  const int lane = threadIdx.x & 31;
  const int wave = threadIdx.x >> 5;
  const int row = blockIdx.x * 4 + wave;
  const float* wrow = W2 + (size_t)row * NN;

  float ax0 = 0, ax1 = 0, ax2 = 0, ax3 = 0;
  float as0 = 0, as1 = 0, as2 = 0, as3 = 0;

  for (int k = lane * 4; k < NN; k += 128) {
    const v4f w = __builtin_nontemporal_load((const v4f*)(wrow + k));
    const float wq[4] = {w.x, w.y, w.z, w.w};
#pragma unroll
    for (int q = 0; q < 4; ++q) {
      const v4f rx = *(const v4f*)(XaF + (size_t)(k + q) * 4);
      const v4f rs = *(const v4f*)(SaF + (size_t)(k + q) * 4);
      ax0 = fmaf(wq[q], rx.x, ax0); ax1 = fmaf(wq[q], rx.y, ax1);
      ax2 = fmaf(wq[q], rx.z, ax2); ax3 = fmaf(wq[q], rx.w, ax3);
      as0 = fmaf(wq[q], rs.x, as0); as1 = fmaf(wq[q], rs.y, as1);
      as2 = fmaf(wq[q], rs.z, as2); as3 = fmaf(wq[q], rs.w, as3);
    }
  }
#pragma unroll
  for (int off = 16; off > 0; off >>= 1) {
    ax0 += __shfl_xor(ax0, off); ax1 += __shfl_xor(ax1, off);
    ax2 += __shfl_xor(ax2, off); ax3 += __shfl_xor(ax3, off);
    as0 += __shfl_xor(as0, off); as1 += __shfl_xor(as1, off);
    as2 += __shfl_xor(as2, off); as3 += __shfl_xor(as3, off);
  }
  if (lane == 0) {
    MxF[row * 4 + 0] = ax0; MxF[row * 4 + 1] = ax1;
    MxF[row * 4 + 2] = ax2; MxF[row * 4 + 3] = ax3;
    MsF[row * 4 + 0] = as0; MsF[row * 4 + 1] = as1;
    MsF[row * 4 + 2] = as2; MsF[row * 4 + 3] = as3;
  }
}

// Epilogue: Xm/Sm elementwise, +sin(k) on the S branch, degenerate 3x3 SAME
// conv (B=1 -> only kw=center taps), write two (8, 8192) slabs at out and out+524288.
__global__ __launch_bounds__(128) void k_final(
    const float* __restrict__ X, const float* __restrict__ S,
    const float* __restrict__ MxF, const float* __restrict__ MsF,
    const float* __restrict__ cwx, const float* __restrict__ cbx,
    const float* __restrict__ cws, const float* __restrict__ cbs,
    float* __restrict__ out) {
  const int h = blockIdx.x * 128 + threadIdx.x;
  float inx[3][8], ins[3][8];
#pragma unroll
  for (int d = 0; d < 3; ++d) {
    const int hh = h + d - 1;
    const bool ok = (hh >= 0) && (hh < NN);
    const float sv = ok ? __sinf((float)hh) : 0.0f;  // sinusoid table, d_hid=1
#pragma unroll
    for (int c = 0; c < 8; ++c) {
      float xv = 0.f, svv = 0.f;
      if (ok) {
        if (c < 4) {
          const float xb = X[c * NN + hh];
          xv  = MxF[c * NN + hh] * xb;           // Xm
          svv = MsF[c * NN + hh] * xb + sv;      // Sm (ref multiplies by X)
        } else {
          xv  = X[(c - 4) * NN + hh];
          svv = S[(c - 4) * NN + hh] + sv;
        }
      }
      inx[d][c] = xv;
      ins[d][c] = svv;
    }
  }
#pragma unroll
  for (int co = 0; co < 8; ++co) {
    float ax = cbx[co], as = cbs[co];
#pragma unroll
    for (int ci = 0; ci < 8; ++ci) {
#pragma unroll
      for (int d = 0; d < 3; ++d) {
        const int wi = ((co * 8 + ci) * 3 + d) * 3 + 1;  // [co,ci,kh,kw=1]
        ax = fmaf(cwx[wi], inx[d][ci], ax);
        as = fmaf(cws[wi], ins[d][ci], as);
      }
    }
    out[co * NN + h] = ax;                 // X_e_hat
    out[8 * NN * 8 + co * NN + h] = as;    // S_e_hat (offset 524288)
  }
}

extern "C" void kernel_launch(void* const* d_in, const int* in_sizes, int n_in,
                              void* d_out, int out_size, void* d_ws,
                              size_t ws_size, hipStream_t stream) {
  const float* X   = (const float*)d_in[0];
  const float* S   = (const float*)d_in[1];
  const float* W1x = (const float*)d_in[2];
  const float* W1s = (const float*)d_in[3];
  const float* W2x = (const float*)d_in[4];
  const float* cwx = (const float*)d_in[5];
  const float* cbx = (const float*)d_in[6];
  const float* cws = (const float*)d_in[7];
  const float* cbs = (const float*)d_in[8];
  float* out = (float*)d_out;

  float* ws = (float*)d_ws;
  const size_t PLANE = (size_t)LL * NN;   // 32768 floats
  float* bounds = ws;                     // 16 floats reserved
  float* Xe = ws + 16;
  float* Se = Xe + PLANE;
  float* Xa = Se + PLANE;
  float* Sa = Xa + PLANE;
  float* Mx = Sa + PLANE;
  float* Ms = Mx + PLANE;

  k_init<<<1, 32, 0, stream>>>(bounds);
  k_proj<<<NN / 128, 128, 0, stream>>>(X, S, W1x, W1s, Xe, Se, bounds);
  k_attn<<<dim3(NN / 64, 2), 128, 0, stream>>>(Xe, Se, Xa, Sa, bounds);
  k_gemm2<<<NN / 4, 128, 0, stream>>>(W2x, Xa, Sa, Mx, Ms);
  k_final<<<NN / 128, 128, 0, stream>>>(X, S, Mx, Ms, cwx, cbx, cws, cbs, out);
}